// StructNDeconv2D_gy_with_ds_34505767256403
// MI455X (gfx1250) — compile-verified
//
#include <hip/hip_runtime.h>

// Problem constants (match the reference)
#define KB 4          // batch
#define KC 16         // channels
#define KH 128
#define KW 128
#define KOH 255
#define KOW 255
#define KEPS 1e-20f

#define N_SRC (KB * KC * KH * KW)        // 1,048,576
#define N_OUT (KB * KC * KOH * KOW)      // 4,161,600
#define PLANE_OUT (KOH * KOW)            // 65,025
#define N_ROWS (KB * KC * KOH)           // 16,320 output rows

typedef float vfloat4 __attribute__((ext_vector_type(4)));
typedef vfloat4 vfloat4u __attribute__((aligned(4)));   // 4B-aligned vector view

#if defined(__gfx1250__) && __has_builtin(__builtin_amdgcn_global_load_async_to_lds_b32)
#define HAVE_ASYNC_LDS 1
typedef __attribute__((address_space(1))) int ga_int;   // global AS pointer pointee
typedef __attribute__((address_space(3))) int ls_int;   // LDS AS pointer pointee
#endif

// ---------------------------------------------------------------------------
// Fused per-source-pixel computation of (gy_f, cgy_f).
// Faithful to the reference: d/s roll with wrap; cd_up row0 / cd_down rowH-1
// forced to zero; cd_up used twice in cgy_from_ds (as in source).
// ---------------------------------------------------------------------------
__device__ __forceinline__ float2 compute_gyf(
    const float* __restrict__ d, const float* __restrict__ cd,
    const float* __restrict__ s, const float* __restrict__ gy,
    const float* __restrict__ cgy, float wp,
    int bc, int y, int x)
{
    const int plane = bc * (KH * KW);
    const int idx   = plane + y * KW + x;
    const int ym1   = (y == 0)      ? (KH - 1) : (y - 1);   // jnp.roll wraps
    const int yp1   = (y == KH - 1) ? 0        : (y + 1);
    const int iu    = plane + ym1 * KW + x;
    const int idn   = plane + yp1 * KW + x;

    const float s_c  = s[idx];
    const float s_u  = s[iu];
    const float s_d  = s[idn];
    const float d_u  = d[iu];
    const float d_d  = d[idn];
    const float cd_u = (y == 0)      ? 0.0f : cd[iu];
    const float cd_d = (y == KH - 1) ? 0.0f : cd[idn];

    const float cgy_from_ds = s_c * s_u * s_d * cd_u * cd_u;      // cd_up twice
    const float height      = (cd_u * d_u + cd_d * d_d) / (cd_u + cd_d);
    const float gy_from_ds  = ((d_d - d_u) * 0.5f) / height;

    const float wc    = wp * cgy[idx];
    const float gy_f  = (wc * gy[idx] + cgy_from_ds * gy_from_ds) / (wc + cgy_from_ds);
    const float cgy_f = (wc + cgy_from_ds) / (wp + 1.0f);
    return make_float2(gy_f, cgy_f);
}

// ---------------------------------------------------------------------------
// Pass 1: stage gy_f / cgy_f (8 MB total) into workspace; stays L2-resident.
// ---------------------------------------------------------------------------
__global__ void __launch_bounds__(256)
pass1_gyf(const float* __restrict__ d,  const float* __restrict__ cd,
          const float* __restrict__ s,  const float* __restrict__ gy,
          const float* __restrict__ cgy, const float* __restrict__ wprop,
          float* __restrict__ gyf, float* __restrict__ cgyf)
{
    const int idx = blockIdx.x * blockDim.x + threadIdx.x;
    if (idx >= N_SRC) return;
    const int x  = idx & (KW - 1);
    const int y  = (idx >> 7) & (KH - 1);
    const int bc = idx >> 14;
    const float2 r = compute_gyf(d, cd, s, gy, cgy, wprop[0], bc, y, x);
    gyf[idx]  = r.x;
    cgyf[idx] = r.y;
}

// ---------------------------------------------------------------------------
// Pass 2 (vectorized): each thread computes 4 consecutive ow pixels of one
// output row.  Block = 256 threads = 4 rows x 64 threads (64*4 = 256 >= 255).
// s_prod_roll rows are read as b128 NT loads (parity-valid m-planes only);
// outputs written as b128 NT stores.  gy_f/cgy_f gathers (L2 hits) are reused
// across the 4 pixels via registers.  spatial_weight/bias staged into LDS via
// the gfx1250 async global->LDS path when available.
// ---------------------------------------------------------------------------
__global__ void __launch_bounds__(256)
pass2_deconv(const float* __restrict__ sp,    // s_prod_roll  (B,C,9,255,255)
             const float* __restrict__ sw,    // spatial_weight flat (C,9)
             const float* __restrict__ bias,  // (C)
             const float* __restrict__ gyf,
             const float* __restrict__ cgyf,
             float* __restrict__ out0, float* __restrict__ out1)
{
    __shared__ float lsw[KC * 9];
    __shared__ float lbias[KC];

    const int t   = threadIdx.x;
    const int r   = blockIdx.x * 4 + (t >> 6);   // output row id, < 16320 always
    const int q   = t & 63;
    const int ow0 = q * 4;                       // 0..252, always even
    const int bc  = r / KOH;
    const int oh  = r % KOH;
    const int c   = bc & (KC - 1);
    const int plane = bc * (KH * KW);
    const int xs0 = ow0 >> 1;
    const bool full = (ow0 != 252);              // 4 valid pixels vs 3 (tail)

    // Warm the gather row (L2-resident scratch) while LDS staging completes.
    __builtin_prefetch(&gyf[plane + (oh >> 1) * KW + xs0], 0, 3);

    // --- stage (C,9) weights + bias into LDS -------------------------------
#if defined(HAVE_ASYNC_LDS)
    if (t < KC * 9)
        __builtin_amdgcn_global_load_async_to_lds_b32(
            (ga_int*)(sw + t), (ls_int*)&lsw[t], 0, 0);
    if (t < KC)
        __builtin_amdgcn_global_load_async_to_lds_b32(
            (ga_int*)(bias + t), (ls_int*)&lbias[t], 0, 0);
#if __has_builtin(__builtin_amdgcn_s_wait_asynccnt)
    __builtin_amdgcn_s_wait_asynccnt(0);
#else
    asm volatile("s_wait_asynccnt 0x0" ::: "memory");
#endif
#else
    if (t < KC * 9) lsw[t]  = sw[t];
    if (t < KC)     lbias[t] = bias[t];
#endif
    __syncthreads();

    // Parity-valid kernel rows: oh even -> i=1 ; oh odd -> i in {0,2}.
    int ilist[2]; int ni;
    if ((oh & 1) == 0) { ni = 1; ilist[0] = 1; ilist[1] = 1; }
    else               { ni = 2; ilist[0] = 0; ilist[1] = 2; }

    float nom[4] = {0.f, 0.f, 0.f, 0.f};
    float den[4] = {0.f, 0.f, 0.f, 0.f};
    float cdenE = 0.f, cdenO = 0.f;

    #pragma unroll
    for (int a = 0; a < 2; ++a) {
        if (a >= ni) break;
        const int i    = ilist[a];
        const int ys   = (oh + 1 - i) >> 1;
        const int srow = plane + ys * KW;

        // Source columns xs0..xs0+2 cover all 4 pixels' taps (register reuse).
        const float cg0 = cgyf[srow + xs0];
        const float g0  = gyf [srow + xs0];
        const float cg1 = cgyf[srow + xs0 + 1];
        const float g1  = gyf [srow + xs0 + 1];
        float cg2 = 0.0f, g2 = 0.0f;
        if (full) {                       // xs0+2 only exists for pixel e=3
            cg2 = cgyf[srow + xs0 + 2];
            g2  = gyf [srow + xs0 + 2];
        }

        // Three m-planes (j=0,1,2) for this kernel row; streamed once -> NT.
        const int mb = (bc * 9 + i * 3) * PLANE_OUT + oh * KOW + ow0;
        vfloat4 sp0, sp1, sp2;
        if (full) {
            sp0 = __builtin_nontemporal_load((const vfloat4u*)&sp[mb]);
            sp1 = __builtin_nontemporal_load((const vfloat4u*)&sp[mb + PLANE_OUT]);
            sp2 = __builtin_nontemporal_load((const vfloat4u*)&sp[mb + 2 * PLANE_OUT]);
        } else {                          // tail: avoid 4B overrun past buffers
            sp0.x = __builtin_nontemporal_load(&sp[mb]);
            sp0.y = __builtin_nontemporal_load(&sp[mb + 1]);
            sp0.z = __builtin_nontemporal_load(&sp[mb + 2]);
            sp0.w = 0.f;
            sp1.x = __builtin_nontemporal_load(&sp[mb + PLANE_OUT]);
            sp1.y = __builtin_nontemporal_load(&sp[mb + PLANE_OUT + 1]);
            sp1.z = __builtin_nontemporal_load(&sp[mb + PLANE_OUT + 2]);
            sp1.w = 0.f;
            sp2.x = __builtin_nontemporal_load(&sp[mb + 2 * PLANE_OUT]);
            sp2.y = __builtin_nontemporal_load(&sp[mb + 2 * PLANE_OUT + 1]);
            sp2.z = __builtin_nontemporal_load(&sp[mb + 2 * PLANE_OUT + 2]);
            sp2.w = 0.f;
        }

        const float w0 = lsw[c * 9 + i * 3 + 0];
        const float w1 = lsw[c * 9 + i * 3 + 1];
        const float w2 = lsw[c * 9 + i * 3 + 2];

        float cp;
        // e=0 (even ow): tap j=1, source col k=0
        cp = cg0 * sp1.x;  nom[0] += w1 * (cp * g0);  den[0] += w1 * cp;
        // e=1 (odd ow):  tap j=0 (k=1), tap j=2 (k=0)
        cp = cg1 * sp0.y;  nom[1] += w0 * (cp * g1);  den[1] += w0 * cp;
        cp = cg0 * sp2.y;  nom[1] += w2 * (cp * g0);  den[1] += w2 * cp;
        // e=2 (even ow): tap j=1, source col k=1
        cp = cg1 * sp1.z;  nom[2] += w1 * (cp * g1);  den[2] += w1 * cp;
        // e=3 (odd ow):  tap j=0 (k=2), tap j=2 (k=1)   (zeros when !full)
        cp = cg2 * sp0.w;  nom[3] += w0 * (cp * g2);  den[3] += w0 * cp;
        cp = cg1 * sp2.w;  nom[3] += w2 * (cp * g1);  den[3] += w2 * cp;

        cdenE += w1;
        cdenO += w0 + w2;
    }

    const float bv = lbias[c];
    vfloat4 o0v, o1v;
    #pragma unroll
    for (int e = 0; e < 4; ++e) {
        float go = nom[e] / (den[e] + KEPS);
        go = go + bv;          // bias added twice, as in source
        go = go + bv;
        go = go * 0.5f;        // / STRIDE
        const float cd_e = (e & 1) ? cdenO : cdenE;
        const float o1   = den[e] / (cd_e + KEPS);
        if (e == 0) { o0v.x = go; o1v.x = o1; }
        if (e == 1) { o0v.y = go; o1v.y = o1; }
        if (e == 2) { o0v.z = go; o1v.z = o1; }
        if (e == 3) { o0v.w = go; o1v.w = o1; }
    }

    const int o = r * KOW + ow0;
    if (full) {
        __builtin_nontemporal_store(o0v, (vfloat4u*)&out0[o]);
        __builtin_nontemporal_store(o1v, (vfloat4u*)&out1[o]);
    } else {
        __builtin_nontemporal_store(o0v.x, &out0[o]);
        __builtin_nontemporal_store(o0v.y, &out0[o + 1]);
        __builtin_nontemporal_store(o0v.z, &out0[o + 2]);
        __builtin_nontemporal_store(o1v.x, &out1[o]);
        __builtin_nontemporal_store(o1v.y, &out1[o + 1]);
        __builtin_nontemporal_store(o1v.z, &out1[o + 2]);
    }
}

// ---------------------------------------------------------------------------
// Fallback: fully fused (no workspace) — recomputes gy_f/cgy_f per tap.
// All recompute loads are L2 hits (20 MB working set << 192 MB L2).
// ---------------------------------------------------------------------------
__global__ void __launch_bounds__(256)
fused_deconv(const float* __restrict__ d,  const float* __restrict__ cd,
             const float* __restrict__ s,  const float* __restrict__ gy,
             const float* __restrict__ cgy, const float* __restrict__ wprop,
             const float* __restrict__ sp, const float* __restrict__ sw,
             const float* __restrict__ bias,
             float* __restrict__ out0, float* __restrict__ out1)
{
    const int idx = blockIdx.x * blockDim.x + threadIdx.x;
    if (idx >= N_OUT) return;

    const int ow = idx % KOW;
    const int t  = idx / KOW;
    const int oh = t % KOH;
    const int bc = t / KOH;
    const int c  = bc & (KC - 1);
    const float wp = wprop[0];

    int ilist[2]; int ni;
    if ((oh & 1) == 0) { ni = 1; ilist[0] = 1; ilist[1] = 1; }
    else               { ni = 2; ilist[0] = 0; ilist[1] = 2; }
    int jlist[2]; int nj;
    if ((ow & 1) == 0) { nj = 1; jlist[0] = 1; jlist[1] = 1; }
    else               { nj = 2; jlist[0] = 0; jlist[1] = 2; }

    const int sp_base = bc * 9 * PLANE_OUT + oh * KOW + ow;

    float nom = 0.0f, den = 0.0f, cden = 0.0f;
    #pragma unroll
    for (int a = 0; a < 2; ++a) {
        if (a >= ni) break;
        const int i  = ilist[a];
        const int ys = (oh + 1 - i) >> 1;
        #pragma unroll
        for (int b = 0; b < 2; ++b) {
            if (b >= nj) break;
            const int j  = jlist[b];
            const int xs = (ow + 1 - j) >> 1;
            const int m  = i * 3 + j;
            const float w   = sw[c * 9 + m];
            const float spv = __builtin_nontemporal_load(&sp[sp_base + m * PLANE_OUT]);
            const float2 rr = compute_gyf(d, cd, s, gy, cgy, wp, bc, ys, xs);
            const float cp  = rr.y * spv;
            nom  += w * (cp * rr.x);
            den  += w * cp;
            cden += w;
        }
    }

    float g_out = nom / (den + KEPS);
    const float bv = bias[c];
    g_out = g_out + bv;
    g_out = g_out + bv;
    g_out = g_out * 0.5f;

    __builtin_nontemporal_store(g_out,               &out0[idx]);
    __builtin_nontemporal_store(den / (cden + KEPS), &out1[idx]);
}

// ---------------------------------------------------------------------------
// Host launcher.  Input dict order:
// 0:d 1:cd 2:s 3:cs 4:gx 5:cgx 6:gy 7:cgy 8:s_prod_roll 9:w_prop
// 10:spatial_weight 11:bias   (cs/gx/cgx unused by the forward pass)
// Output: concat(gy_out/STRIDE, cgy_out), each (B,C,255,255) f32.
// ---------------------------------------------------------------------------
extern "C" void kernel_launch(void* const* d_in, const int* in_sizes, int n_in,
                              void* d_out, int out_size, void* d_ws, size_t ws_size,
                              hipStream_t stream) {
    (void)in_sizes; (void)n_in; (void)out_size;
    const float* d     = (const float*)d_in[0];
    const float* cd    = (const float*)d_in[1];
    const float* s     = (const float*)d_in[2];
    const float* gy    = (const float*)d_in[6];
    const float* cgy   = (const float*)d_in[7];
    const float* sp    = (const float*)d_in[8];
    const float* wprop = (const float*)d_in[9];
    const float* sw    = (const float*)d_in[10];
    const float* bias  = (const float*)d_in[11];

    float* out0 = (float*)d_out;
    float* out1 = out0 + (size_t)N_OUT;

    const int threads = 256;
    const size_t need_ws = 2ull * N_SRC * sizeof(float);

    if (ws_size >= need_ws) {
        float* gyf  = (float*)d_ws;
        float* cgyf = gyf + N_SRC;
        pass1_gyf<<<(N_SRC + threads - 1) / threads, threads, 0, stream>>>(
            d, cd, s, gy, cgy, wprop, gyf, cgyf);
        pass2_deconv<<<N_ROWS / 4, threads, 0, stream>>>(
            sp, sw, bias, gyf, cgyf, out0, out1);
    } else {
        fused_deconv<<<(N_OUT + threads - 1) / threads, threads, 0, stream>>>(
            d, cd, s, gy, cgy, wprop, sp, sw, bias, out0, out1);
    }
}